// SAGEAutoencoder_4827543241246
// MI455X (gfx1250) — compile-verified
//
#include <hip/hip_runtime.h>
#include <hip/hip_bf16.h>

typedef __attribute__((ext_vector_type(2))) float v2f;
typedef __attribute__((ext_vector_type(8))) float v8f;

#define NMEM 50000
#define NPROV 50000
#define NNODES 100000
#define DMEM 96
#define DPROV 128
#define INDIM 128
#define HID 256
#define LAT 64
#define NEDGE 1000000

// ---------------- build padded/concatenated feature matrix X [NNODES, 128] ----
__global__ void build_x_kernel(const float* __restrict__ xm,
                               const float* __restrict__ xp,
                               float* __restrict__ X) {
    long gid = (long)blockIdx.x * blockDim.x + threadIdx.x;
    if (gid >= (long)NNODES * INDIM) return;
    int row = (int)(gid >> 7);
    int col = (int)(gid & 127);
    float v;
    if (row < NMEM) v = (col < DMEM) ? xm[(long)row * DMEM + col] : 0.0f;
    else            v = xp[(long)(row - NMEM) * DPROV + col];
    X[gid] = v;
}

// ---------------- degree (as float) -------------------------------------------
__global__ void degree_kernel(const int* __restrict__ prov,
                              const int* __restrict__ memb,
                              float* __restrict__ deg) {
    int e = blockIdx.x * blockDim.x + threadIdx.x;
    if (e >= 2 * NEDGE) return;
    int dst = (e < NEDGE) ? (memb[e] + NMEM) : prov[e - NEDGE];
    atomicAdd(&deg[dst], 1.0f);
}

__global__ void rdeg_kernel(float* __restrict__ deg) {
    int i = blockIdx.x * blockDim.x + threadIdx.x;
    if (i < NNODES) deg[i] = 1.0f / fmaxf(deg[i], 1.0f);
}

// ---------------- scatter-add of source features into agg ---------------------
// D floats per node; each thread handles one (edge, 4-float chunk).
template <int D, int LOG2_CPE>
__global__ void scatter_kernel(const int* __restrict__ prov,
                               const int* __restrict__ memb,
                               const float* __restrict__ in,
                               float* __restrict__ out) {
    long gid = (long)blockIdx.x * blockDim.x + threadIdx.x;
    if (gid >= (long)2 * NEDGE * (D / 4)) return;
    int  c = (int)(gid & ((D / 4) - 1));
    long e = gid >> LOG2_CPE;
    int src, dst;
    if (e < NEDGE) { src = prov[e];             dst = memb[e] + NMEM; }
    else           { src = memb[e - NEDGE] + NMEM; dst = prov[e - NEDGE]; }
    const float4 v = *(const float4*)(in + (long)src * D + c * 4);
    float* o = out + (long)dst * D + c * 4;
    atomicAdd(o + 0, v.x);
    atomicAdd(o + 1, v.y);
    atomicAdd(o + 2, v.z);
    atomicAdd(o + 3, v.w);
}

// ---------------- WMMA GEMM: out[16r x 64c tile] -------------------------------
// out = act( (A0 * scale0_row) @ W0^T  [+ A1 @ W1^T]  + bias )
// W row-major [N, K]. One wave (32 threads) per block; exact tiling, EXEC all-1s.
__device__ __forceinline__ v2f ld2(const float* p) {
    v2f r; r.x = p[0]; r.y = p[1]; return r;
}

__global__ void sage_gemm_wmma(const float* __restrict__ A0,
                               const float* __restrict__ scale0, // 1/deg or null
                               const float* __restrict__ W0,
                               const float* __restrict__ A1,     // may be null
                               const float* __restrict__ W1,     // may be null
                               const float* __restrict__ bias,
                               float* __restrict__ out,
                               int K, int N, int relu) {
    const int lane = threadIdx.x;
    const int l16  = lane & 15;
    const int hi   = lane >> 4;           // 0: K-pair {0,1}, 1: K-pair {2,3}
    const int row0 = blockIdx.x * 16;
    const int col0 = blockIdx.y * 64;

    v8f acc[4];
    #pragma unroll
    for (int t = 0; t < 4; ++t) acc[t] = (v8f){};

    const int   arow = row0 + l16;        // A lane row (A layout: M = lane%16)
    const float s0   = scale0 ? scale0[arow] : 1.0f;

    // phase 0: scaled aggregate term
    {
        const float* Ar = A0 + (long)arow * K;
        for (int k = 0; k < K; k += 4) {
            const int ko = k + 2 * hi;
            v2f a; a.x = Ar[ko] * s0; a.y = Ar[ko + 1] * s0;
            #pragma unroll
            for (int t = 0; t < 4; ++t) {
                const int n = col0 + t * 16 + l16;
                v2f b = ld2(W0 + (long)n * K + ko);
                acc[t] = __builtin_amdgcn_wmma_f32_16x16x4_f32(
                    false, a, false, b, (short)0, acc[t], false, false);
            }
        }
    }
    // phase 1: self term (x @ Wr^T)
    if (A1) {
        const float* Ar = A1 + (long)arow * K;
        for (int k = 0; k < K; k += 4) {
            const int ko = k + 2 * hi;
            v2f a; a.x = Ar[ko]; a.y = Ar[ko + 1];
            #pragma unroll
            for (int t = 0; t < 4; ++t) {
                const int n = col0 + t * 16 + l16;
                v2f b = ld2(W1 + (long)n * K + ko);
                acc[t] = __builtin_amdgcn_wmma_f32_16x16x4_f32(
                    false, a, false, b, (short)0, acc[t], false, false);
            }
        }
    }
    // epilogue: C/D layout -> VGPR r holds M = r + 8*hi, N = lane%16
    #pragma unroll
    for (int t = 0; t < 4; ++t) {
        const int col = col0 + t * 16 + l16;
        const float bv = bias[col];
        #pragma unroll
        for (int r = 0; r < 8; ++r) {
            const int row = row0 + r + 8 * hi;
            float v = acc[t][r] + bv;
            if (relu) v = fmaxf(v, 0.0f);
            out[(long)row * N + col] = v;
        }
    }
}

// ---------------- edge logits: dot(z_prov, z_mem) over 64 dims ------------------
__global__ void edge_logits_kernel(const int* __restrict__ prov,
                                   const int* __restrict__ memb,
                                   const float* __restrict__ z,
                                   float* __restrict__ out) {
    int e = blockIdx.x * blockDim.x + threadIdx.x;
    if (e >= NEDGE) return;
    const float* zp = z + (long)(NMEM + prov[e]) * LAT;
    const float* zm = z + (long)memb[e] * LAT;
    float s = 0.0f;
    #pragma unroll
    for (int j = 0; j < LAT; j += 4) {
        float4 a = *(const float4*)(zp + j);
        float4 b = *(const float4*)(zm + j);
        s += a.x * b.x + a.y * b.y + a.z * b.z + a.w * b.w;
    }
    out[e] = s;
}

extern "C" void kernel_launch(void* const* d_in, const int* in_sizes, int n_in,
                              void* d_out, int out_size, void* d_ws, size_t ws_size,
                              hipStream_t stream) {
    (void)in_sizes; (void)n_in; (void)out_size; (void)ws_size;
    const float* xm  = (const float*)d_in[0];
    const float* xp  = (const float*)d_in[1];
    const int*   ei  = (const int*)d_in[2];   // [2, E]
    const float* W1l = (const float*)d_in[3];
    const float* W1r = (const float*)d_in[4];
    const float* b1  = (const float*)d_in[5];
    const float* W2l = (const float*)d_in[6];
    const float* W2r = (const float*)d_in[7];
    const float* b2  = (const float*)d_in[8];
    const float* Wd  = (const float*)d_in[9];
    const float* bd  = (const float*)d_in[10];
    const int* prov = ei;
    const int* memb = ei + NEDGE;

    float* ws = (float*)d_ws;
    // workspace layout (floats); AGG2 aliases X+AGG1 (dead after layer-1 GEMM)
    const size_t O_RDEG = 0;
    const size_t O_Z    = 131072;
    const size_t O_H    = O_Z + (size_t)NNODES * LAT;
    const size_t O_X    = O_H + (size_t)NNODES * HID;
    const size_t O_AGG1 = O_X + (size_t)NNODES * INDIM;
    float* rdeg = ws + O_RDEG;
    float* Z    = ws + O_Z;
    float* H    = ws + O_H;
    float* X    = ws + O_X;
    float* AGG1 = ws + O_AGG1;
    float* AGG2 = ws + O_X;   // alias: spans X+AGG1 regions (NNODES*HID floats)

    float* xhat   = (float*)d_out;                          // [100000,128] (member rows first)
    float* logits = (float*)d_out + (size_t)NNODES * INDIM; // [1e6]

    hipMemsetAsync(rdeg, 0, (size_t)NNODES * sizeof(float), stream);
    hipMemsetAsync(AGG1, 0, (size_t)NNODES * INDIM * sizeof(float), stream);

    { long tot = (long)NNODES * INDIM;
      build_x_kernel<<<(unsigned)((tot + 255) / 256), 256, 0, stream>>>(xm, xp, X); }

    degree_kernel<<<(2 * NEDGE + 255) / 256, 256, 0, stream>>>(prov, memb, rdeg);
    rdeg_kernel<<<(NNODES + 255) / 256, 256, 0, stream>>>(rdeg);

    { long tot = (long)2 * NEDGE * (INDIM / 4);
      scatter_kernel<INDIM, 5><<<(unsigned)((tot + 255) / 256), 256, 0, stream>>>(prov, memb, X, AGG1); }

    dim3 g1(NNODES / 16, HID / 64);
    sage_gemm_wmma<<<g1, 32, 0, stream>>>(AGG1, rdeg, W1l, X, W1r, b1, H, INDIM, HID, 1);

    hipMemsetAsync(AGG2, 0, (size_t)NNODES * HID * sizeof(float), stream);
    { long tot = (long)2 * NEDGE * (HID / 4);
      scatter_kernel<HID, 6><<<(unsigned)((tot + 255) / 256), 256, 0, stream>>>(prov, memb, H, AGG2); }

    dim3 g2(NNODES / 16, LAT / 64);
    sage_gemm_wmma<<<g2, 32, 0, stream>>>(AGG2, rdeg, W2l, H, W2r, b2, Z, HID, LAT, 0);

    dim3 g3(NNODES / 16, INDIM / 64);
    sage_gemm_wmma<<<g3, 32, 0, stream>>>(Z, nullptr, Wd, nullptr, nullptr, bd, xhat, LAT, INDIM, 0);

    edge_logits_kernel<<<(NEDGE + 255) / 256, 256, 0, stream>>>(prov, memb, Z, logits);
}